// SinkNodeAttentionBias_69200513073395
// MI455X (gfx1250) — compile-verified
//
#include <hip/hip_runtime.h>
#include <hip/hip_bf16.h>

// Problem constants fixed by the reference: B=4, L=2048, H=16.
// Output: bias[B,H,L,L] float32 = 1 GiB, zero everywhere except [:, :, 0, :].
// Pure store-bandwidth problem: 1 GiB / 23.3 TB/s ~= 46 us floor.

typedef float v4f __attribute__((ext_vector_type(4)));
typedef int   v4i __attribute__((ext_vector_type(4)));

#ifndef __has_builtin
#define __has_builtin(x) 0
#endif

// CDNA5 async store path: stream zeros LDS -> HBM via GLOBAL_STORE_ASYNC_FROM_LDS_B128
// (tracked with ASYNCcnt, bypasses the VGPR store-data path). Builtin signature
// (learned from the round-1 diagnostic / upstream clang):
//   void __builtin_amdgcn_global_store_async_from_lds_b128(
//        int4 addrspace(1)* gdst, int4 addrspace(3)* lsrc, imm int off, imm int cpol)
#if __has_builtin(__builtin_amdgcn_global_store_async_from_lds_b128)
#define USE_ASYNC_STORE 1
#else
#define USE_ASYNC_STORE 0
#endif

constexpr int kB = 4;
constexpr int kL = 2048;
constexpr int kH = 16;

// Address-space-qualified int4 pointee types for the async builtin.
typedef __attribute__((address_space(1))) v4i gv4i;  // global
typedef __attribute__((address_space(3))) v4i lv4i;  // LDS

// ---------------------------------------------------------------------------
// Kernel 1: zero-fill the full 1 GiB output with b128 stores.
// ---------------------------------------------------------------------------
__global__ void __launch_bounds__(256)
sink_bias_zero_fill(float* __restrict__ out, size_t n4 /* # of float4 */) {
  const size_t tid    = (size_t)blockIdx.x * blockDim.x + threadIdx.x;
  const size_t stride = (size_t)gridDim.x * blockDim.x;

#if USE_ASYNC_STORE
  // 16 bytes of zeroed LDS per lane; async stores read LDS directly.
  __shared__ float lds_zero[256 * 4];
  {
    v4f z = {0.0f, 0.0f, 0.0f, 0.0f};
    *(v4f*)&lds_zero[threadIdx.x * 4] = z;
  }
  __syncthreads();  // DS writes visible before async engine reads LDS

  lv4i* lsrc = (lv4i*)(v4i*)(void*)&lds_zero[threadIdx.x * 4];
  for (size_t i = tid; i < n4; i += stride) {
    gv4i* gdst = (gv4i*)(v4i*)(void*)(out + i * 4);
    // 16B per lane per instruction; up to 63 outstanding on ASYNCcnt.
    __builtin_amdgcn_global_store_async_from_lds_b128(gdst, lsrc, 0, 0);
  }
#if __has_builtin(__builtin_amdgcn_s_wait_asynccnt)
  __builtin_amdgcn_s_wait_asynccnt(0);
#else
  asm volatile("s_wait_asynccnt 0x0" ::: "memory");
#endif
#else
  // Fallback: nontemporal b128 stores (global_store_b128 th:NT).
  v4f z = {0.0f, 0.0f, 0.0f, 0.0f};
  v4f* __restrict__ out4 = (v4f*)out;
  for (size_t i = tid; i < n4; i += stride) {
    __builtin_nontemporal_store(z, out4 + i);
  }
#endif
}

// ---------------------------------------------------------------------------
// Kernel 2: compute and write the nonzero row-0 slices: bias[b,h,0,k].
// B*H*L = 131072 elements (512 KB). Recomputing powf per head (x16) is noise.
// ---------------------------------------------------------------------------
__global__ void __launch_bounds__(256)
sink_bias_row_write(const float* __restrict__ c_sink,          // [B*L]
                    const unsigned char* __restrict__ mask,    // [B*L] bool
                    const float* __restrict__ beta_p,
                    const float* __restrict__ floor_p,
                    const float* __restrict__ gamma_p,
                    float* __restrict__ out) {
  const int idx = blockIdx.x * blockDim.x + threadIdx.x;   // [0, B*H*L)
  if (idx >= kB * kH * kL) return;

  const int k  = idx & (kL - 1);    // key position
  const int bh = idx >> 11;         // b*H + h
  const int b  = bh >> 4;           // / H

  const float gamma = gamma_p[0];   // uniform -> scalar loads
  const float flr   = floor_p[0];
  const float beta  = beta_p[0];

  float c = c_sink[b * kL + k];
  c = fminf(fmaxf(c, 0.0f), 1.0f);
  float r = powf(c + 1e-6f, gamma);
  r = flr + (1.0f - flr) * r;
  r = mask[b * kL + k] ? 0.0f : r;

  // bias[b][h][0][k]  ->  flat offset bh * L*L + k
  out[(size_t)bh * (size_t)(kL * kL) + (size_t)k] = beta * r;
}

// ---------------------------------------------------------------------------
extern "C" void kernel_launch(void* const* d_in, const int* in_sizes, int n_in,
                              void* d_out, int out_size, void* d_ws, size_t ws_size,
                              hipStream_t stream) {
  // Inputs (setup_inputs order): c_sink f32[B*L], mask bool[B*L],
  // beta f32[1], floor f32[1], gamma f32[1], H int[1] (unused; fixed 16).
  const float*         c_sink = (const float*)d_in[0];
  const unsigned char* mask   = (const unsigned char*)d_in[1];
  const float*         beta   = (const float*)d_in[2];
  const float*         flr    = (const float*)d_in[3];
  const float*         gamma  = (const float*)d_in[4];
  float*               out    = (float*)d_out;

  // 1) Zero the full output (268,435,456 floats = 67,108,864 float4 stores).
  const size_t n4 = (size_t)out_size / 4;
  sink_bias_zero_fill<<<8192, 256, 0, stream>>>(out, n4);

  // 2) Overwrite the B*H*L row-0 elements with the computed sink row.
  const int row_elems = kB * kH * kL;  // 131072
  sink_bias_row_write<<<(row_elems + 255) / 256, 256, 0, stream>>>(
      c_sink, mask, beta, flr, gamma, out);
}